// Attention_6914897346947
// MI455X (gfx1250) — compile-verified
//
#include <hip/hip_runtime.h>
#include <hip/hip_bf16.h>
#include <math.h>

// ---------------------------------------------------------------------------
// Problem constants (from the reference)
// ---------------------------------------------------------------------------
#define BB   32      // batch
#define TT   1       // query time steps
#define SS   8192    // KV cache length
#define DD   3584    // model dim
#define NN   28      // query heads
#define KH   4       // kv heads
#define HH   128     // head dim
#define GG   (NN / KH)   // 7 grouped query heads per kv head
#define HP   (HH + 4)    // padded LDS row stride (16B-aligned, low conflicts)

typedef float v2f __attribute__((ext_vector_type(2)));
typedef float v8f __attribute__((ext_vector_type(8)));
typedef int   v4i __attribute__((vector_size(16)));
typedef __attribute__((address_space(1))) v4i gas_v4i;   // global (AS1)
typedef __attribute__((address_space(3))) v4i las_v4i;   // LDS    (AS3)

// ---------------------------------------------------------------------------
// CDNA5 async load-to-LDS (ASYNCcnt-tracked) with sync fallback.
// Probe-confirmed signature: (int4 AS1*, int4 AS3*, imm offset, imm cpol).
// ---------------------------------------------------------------------------
__device__ __forceinline__ void stage16(const float* __restrict__ g, float* l) {
#if __has_builtin(__builtin_amdgcn_global_load_async_to_lds_b128)
    __builtin_amdgcn_global_load_async_to_lds_b128(
        (gas_v4i*)(void*)g,
        (las_v4i*)(void*)l,
        0, 0);
#else
    *(float4*)l = *(const float4*)g;
#endif
}

template <int N>
__device__ __forceinline__ void wait_asynccnt() {
#if __has_builtin(__builtin_amdgcn_s_wait_asynccnt)
    __builtin_amdgcn_s_wait_asynccnt(N);
#else
    asm volatile("s_wait_asynccnt %0" ::"n"(N) : "memory");
#endif
}

// ---------------------------------------------------------------------------
// Skinny GEMM: C[32 x Ncols] = A[32 x Kdim] * B[Kdim x Ncols] (+ bias[col])
// 128-thread block = 4 waves; each wave owns a 16x16 output tile, block
// covers 64 columns. Both A (16x32) and B (32x64) chunks staged in LDS via
// coalesced float4 loads; WMMA stream is fed purely from ds_load.
// Strides: sA row = 36 floats, sB row = 72 floats -> conflict-free fragment
// reads for both lane halves (khalf bank offsets stay disjoint).
// Two independent accumulator chains break the WMMA D->C serialization.
// ---------------------------------------------------------------------------
#define KC 32

__global__ __launch_bounds__(128)
void gemm32_wmma(const float* __restrict__ A, const float* __restrict__ Bm,
                 const float* __restrict__ bias, float* __restrict__ C,
                 int Kdim, int lda, int ldb, int ldc) {
    __shared__ __align__(16) float sA[16][36];
    __shared__ __align__(16) float sB[KC][72];

    const int tid   = threadIdx.x;      // 0..127
    const int lane  = tid & 31;
    const int wave  = tid >> 5;         // 0..3
    const int rt    = blockIdx.y;       // row tile: rows rt*16 .. rt*16+15
    const int colBase = blockIdx.x * 64;
    const int mrow  = lane & 15;        // M index (A) / N index (B)
    const int khalf = (lane >> 4) << 1; // 0 for lanes 0-15, 2 for lanes 16-31
    const int bn    = wave * 16 + mrow; // column within the 64-col block

    // cooperative-load coordinates
    const int arow = tid >> 3;          // A: 16 rows x 8 float4
    const int acol = (tid & 7) << 2;
    const int br   = tid >> 4;          // B: 16 float4 per row, 8 rows/iter
    const int bc   = (tid & 15) << 2;

    v8f acc0 = {0.f, 0.f, 0.f, 0.f, 0.f, 0.f, 0.f, 0.f};
    v8f acc1 = {0.f, 0.f, 0.f, 0.f, 0.f, 0.f, 0.f, 0.f};

    for (int k0 = 0; k0 < Kdim; k0 += KC) {
        // A chunk: 16x32 floats = 128 float4, one per thread.
        *(float4*)&sA[arow][acol] =
            *(const float4*)(A + (size_t)(rt * 16 + arow) * lda + k0 + acol);
        // B chunk: 32x64 floats = 512 float4, four per thread (coalesced).
        #pragma unroll
        for (int i = 0; i < 4; ++i) {
            const int r = br + 8 * i;
            *(float4*)&sB[r][bc] =
                *(const float4*)(Bm + (size_t)(k0 + r) * ldb + colBase + bc);
        }
        __syncthreads();

        #pragma unroll
        for (int kk = 0; kk < KC; kk += 8) {
            v2f a0, b0, a1, b1;
            a0.x = sA[mrow][kk + khalf];
            a0.y = sA[mrow][kk + khalf + 1];
            b0.x = sB[kk + khalf][bn];
            b0.y = sB[kk + khalf + 1][bn];
            acc0 = __builtin_amdgcn_wmma_f32_16x16x4_f32(
                false, a0, false, b0, (short)0, acc0, false, false);
            a1.x = sA[mrow][kk + 4 + khalf];
            a1.y = sA[mrow][kk + 4 + khalf + 1];
            b1.x = sB[kk + 4 + khalf][bn];
            b1.y = sB[kk + 4 + khalf + 1][bn];
            acc1 = __builtin_amdgcn_wmma_f32_16x16x4_f32(
                false, a1, false, b1, (short)0, acc1, false, false);
        }
        __syncthreads();
    }

    const int colOut = colBase + bn;
    const float bv = bias ? bias[colOut] : 0.f;
    const int baseRow = rt * 16 + ((lane >> 4) << 3); // +0 or +8
    #pragma unroll
    for (int r = 0; r < 8; ++r) {
        C[(size_t)(baseRow + r) * ldc + colOut] = acc0[r] + acc1[r] + bv;
    }
}

// ---------------------------------------------------------------------------
// RoPE, applied in place to a [B x heads x 128] activation buffer.
// ---------------------------------------------------------------------------
__global__ __launch_bounds__(64)
void rope_kernel(float* __restrict__ vec, const int* __restrict__ seg_pos,
                 int heads) {
    const int bh = blockIdx.x;
    const int b  = bh / heads;
    const int i  = threadIdx.x;              // 0..63
    const float pos  = (float)seg_pos[b];
    const float frac = (2.0f * (float)i) / (float)HH;
    const float ts   = __powf(1000000.0f, frac);
    const float ang  = pos / ts;
    float s, c;
    __sincosf(ang, &s, &c);
    float* base = vec + (size_t)bh * HH;
    const float x1 = base[i];
    const float x2 = base[i + 64];
    base[i]      = x1 * c - x2 * s;
    base[i + 64] = x2 * c + x1 * s;
}

// ---------------------------------------------------------------------------
// Attention: one block per (b, kv-head), 256 threads = 8 waves.
// K/V chunks (32 positions x 128 fp32) are double-buffered in LDS via
// GLOBAL_LOAD_ASYNC_TO_LDS_B128: issue chunk i+1, s_wait_asynccnt 8 (in-order
// completion => chunk i landed), barrier, compute. Waves 0..6 each run one
// grouped Q-head with wave32-shuffle online softmax. The cache inputs are
// NOT mutated; the s == end_index row is patched into the LDS tile from the
// freshly projected k_new/v_new (reference dynamic_update_slice, T=1).
// ---------------------------------------------------------------------------
__global__ __launch_bounds__(256)
void attn_kernel(const float* __restrict__ q,      // [B, N, H] (post-RoPE)
                 const float* __restrict__ knew,   // [B, KH, H] (post-RoPE)
                 const float* __restrict__ vnew,   // [B, KH, H]
                 const float* __restrict__ cache_k,// [B, S, KH, H]
                 const float* __restrict__ cache_v,// [B, S, KH, H]
                 const int*   __restrict__ seg_pos,// [B, T]
                 const int*   __restrict__ end_index, // [B]
                 float* __restrict__ qkv)          // [B, N, H]
{
    __shared__ float sQ[GG][HH];
    __shared__ __align__(16) float sK[2][32][HP];
    __shared__ __align__(16) float sV[2][32][HP];

    const int b    = blockIdx.x;
    const int kv   = blockIdx.y;
    const int tid  = threadIdx.x;
    const int lane = tid & 31;
    const int wave = tid >> 5;          // 0..7; waves 0..6 compute

    const int pos  = seg_pos[b];                 // mask: s <= pos
    const int sIdx = end_index[0] % SS;          // cache write position
    const int nChunks = (pos + 32) / 32;         // ceil((pos+1)/32)

    // Stage the 7 query heads of this kv group.
    for (int idx = tid; idx < GG * HH; idx += 256) {
        const int g = idx >> 7;
        const int h = idx & 127;
        sQ[g][h] = q[((size_t)b * NN + (kv * GG + g)) * HH + h];
    }

    auto stage_chunk = [&](int s0, int buf) {
        #pragma unroll
        for (int i = 0; i < 4; ++i) {
            const int idx4 = tid + 256 * i;      // 0..1023 float4 slots
            const int row  = idx4 >> 5;          // 32 float4 per row
            const int c    = (idx4 & 31) << 2;   // float offset within row
            const size_t gbase =
                (((size_t)b * SS + (s0 + row)) * KH + kv) * HH + c;
            stage16(cache_k + gbase, &sK[buf][row][c]);
            stage16(cache_v + gbase, &sV[buf][row][c]);
        }
    };

    stage_chunk(0, 0);                           // prefetch chunk 0

    const float scale = 0.08838834764831845f;    // 128^-0.5
    float m    = -INFINITY;
    float ssum = 0.f;
    float acc[4] = {0.f, 0.f, 0.f, 0.f};         // h = lane + 32*j

    __syncthreads();                             // sQ visible

    for (int c = 0; c < nChunks; ++c) {
        const int s0  = c * 32;
        const int buf = c & 1;

        if (c + 1 < nChunks) {
            stage_chunk(s0 + 32, buf ^ 1);       // prefetch next chunk
            wait_asynccnt<8>();                  // chunk c has landed
        } else {
            wait_asynccnt<0>();
        }
        __syncthreads();

        // Patch the freshly-written cache row (s == sIdx) from k_new/v_new.
        if (sIdx >= s0 && sIdx < s0 + 32) {
            const int prow = sIdx - s0;
            const size_t nbase = ((size_t)b * KH + kv) * HH;
            if (tid < HH)            sK[buf][prow][tid]      = knew[nbase + tid];
            else if (tid < 2 * HH)   sV[buf][prow][tid - HH] = vnew[nbase + tid - HH];
            __syncthreads();
        }

        if (wave < GG) {
            const int s = s0 + lane;
            const bool valid = (s <= pos);

            float sc = 0.f;
            #pragma unroll 4
            for (int h = 0; h < HH; ++h) {
                sc = fmaf(sQ[wave][h], sK[buf][lane][h], sc);
            }
            float e = valid ? sc * scale : -INFINITY;

            float cmax = e;
            #pragma unroll
            for (int off = 16; off > 0; off >>= 1)
                cmax = fmaxf(cmax, __shfl_xor(cmax, off, 32));
            const float nm   = fmaxf(m, cmax);
            const float corr = __expf(m - nm);
            const float p    = valid ? __expf(e - nm) : 0.f;
            float psum = p;
            #pragma unroll
            for (int off = 16; off > 0; off >>= 1)
                psum += __shfl_xor(psum, off, 32);
            ssum = ssum * corr + psum;
            m = nm;
            #pragma unroll
            for (int j = 0; j < 4; ++j) acc[j] *= corr;

            #pragma unroll 8
            for (int ss = 0; ss < 32; ++ss) {
                const float pv = __shfl(p, ss, 32);
                #pragma unroll
                for (int j = 0; j < 4; ++j)
                    acc[j] = fmaf(pv, sV[buf][ss][lane + 32 * j], acc[j]);
            }
        }
        __syncthreads();
    }

    if (wave < GG) {
        const float inv = 1.0f / ssum;
        const size_t obase = ((size_t)b * NN + (kv * GG + wave)) * HH;
        #pragma unroll
        for (int j = 0; j < 4; ++j)
            qkv[obase + lane + 32 * j] = acc[j] * inv;
    }
}

// ---------------------------------------------------------------------------
// Launcher
// ---------------------------------------------------------------------------
extern "C" void kernel_launch(void* const* d_in, const int* in_sizes, int n_in,
                              void* d_out, int out_size, void* d_ws, size_t ws_size,
                              hipStream_t stream) {
    const float* x    = (const float*)d_in[0];   // (B, T, D)
    const int*   seg  = (const int*)  d_in[1];   // (B, T)
    const float* ck   = (const float*)d_in[2];   // (B, S, KH, H)
    const float* cv   = (const float*)d_in[3];   // (B, S, KH, H)
    const int*   eidx = (const int*)  d_in[4];   // (B,)
    // d_in[5] = attn_mask (bool) -- equivalent to s <= seg_pos, handled inline
    const float* wq   = (const float*)d_in[6];   // (D, N, H)
    const float* wk   = (const float*)d_in[7];   // (D, KH, H)
    const float* wv   = (const float*)d_in[8];   // (D, KH, H)
    const float* wo   = (const float*)d_in[9];   // (N, H, D)
    const float* qb   = (const float*)d_in[10];  // (N*H,)
    const float* kb   = (const float*)d_in[11];  // (KH*H,)
    const float* vb   = (const float*)d_in[12];  // (KH*H,)

    float* ws   = (float*)d_ws;
    float* qbuf = ws;                            // B*N*H   = 114688 floats
    float* knew = qbuf + (size_t)BB * NN * HH;   // B*KH*H  =  16384
    float* vnew = knew + (size_t)BB * KH * HH;   //            16384
    float* qkv  = vnew + (size_t)BB * KH * HH;   // B*N*H   = 114688

    const dim3 blk128(128);

    // Q/K/V projections (bias fused), M = 32 batches.
    gemm32_wmma<<<dim3(DD / 64, 2), blk128, 0, stream>>>(
        x, wq, qb, qbuf, DD, DD, NN * HH, NN * HH);
    gemm32_wmma<<<dim3((KH * HH) / 64, 2), blk128, 0, stream>>>(
        x, wk, kb, knew, DD, DD, KH * HH, KH * HH);
    gemm32_wmma<<<dim3((KH * HH) / 64, 2), blk128, 0, stream>>>(
        x, wv, vb, vnew, DD, DD, KH * HH, KH * HH);

    // RoPE on q and k_new (in place).
    rope_kernel<<<BB * NN, 64, 0, stream>>>(qbuf, seg, NN);
    rope_kernel<<<BB * KH, 64, 0, stream>>>(knew, seg, KH);

    // Flash-style attention, one block per (b, kv head).
    attn_kernel<<<dim3(BB, KH), 256, 0, stream>>>(
        qbuf, knew, vnew, ck, cv, seg, eidx, qkv);

    // Output projection: out[b, d] = sum_nh qkv[b, nh] * wo[nh, d].
    gemm32_wmma<<<dim3(DD / 64, 2), blk128, 0, stream>>>(
        qkv, wo, nullptr, (float*)d_out, NN * HH, NN * HH, DD, DD);
}